// SummaryWithAttention_36318243455702
// MI455X (gfx1250) — compile-verified
//
#include <hip/hip_runtime.h>

// out[n,:] = ((sum_s V[n,s,:]) @ Wv.T + 4096*bv) @ Wo.T + bo
// (softmax over a size-1 axis in the reference makes attention weights == 1,
//  so Q/K/Wq/Wk/bq/bk are mathematically dead)

typedef __attribute__((ext_vector_type(2))) float v2f;
typedef __attribute__((ext_vector_type(8))) float v8f;

#define DM   1024
#define NB   8
#define SEQ  4096
#define SCHUNKS 16
#define SPER (SEQ / SCHUNKS)   // 256

// ---------------------------------------------------------------------------
// Kernel 1: partial reduction over S with float4 (global_load_b128) streams.
// grid = (1, SCHUNKS, NB), block = 256; each thread owns 4 contiguous d's.
// partial[(n*SCHUNKS + sc)*DM + 4t .. 4t+3] = sum_{s in chunk} V[n, s, 4t..]
// ---------------------------------------------------------------------------
__global__ void reduceV_partial(const float* __restrict__ V,
                                float* __restrict__ partial) {
    const int t  = threadIdx.x;      // 0..255 -> d = 4t
    const int sc = blockIdx.y;       // 0..15
    const int n  = blockIdx.z;       // 0..7
    const float* p = V + ((size_t)n * SEQ + (size_t)sc * SPER) * DM + 4 * t;
    float4 acc = make_float4(0.f, 0.f, 0.f, 0.f);
    #pragma unroll 4
    for (int s = 0; s < SPER; ++s) {
        __builtin_prefetch(p + 8 * DM, 0, 0);   // global_prefetch_b8, 32KB ahead
        float4 v = *(const float4*)p;
        acc.x += v.x; acc.y += v.y; acc.z += v.z; acc.w += v.w;
        p += DM;
    }
    *(float4*)(partial + ((size_t)n * SCHUNKS + sc) * DM + 4 * t) = acc;
}

// ---------------------------------------------------------------------------
// Kernel 2: combine partials -> zero-padded Vsum16 [16, DM] (rows 8..15 = 0)
// ---------------------------------------------------------------------------
__global__ void combine_partials(const float* __restrict__ partial,
                                 float* __restrict__ vsum16) {
    const int i = blockIdx.x * blockDim.x + threadIdx.x; // 0..16383
    const int row = i >> 10;
    const int d   = i & (DM - 1);
    float acc = 0.0f;
    if (row < NB) {
        #pragma unroll
        for (int c = 0; c < SCHUNKS; ++c)
            acc += partial[((size_t)row * SCHUNKS + c) * DM + d];
    }
    vsum16[i] = acc;
}

// ---------------------------------------------------------------------------
// GEMM: out[m, :] = A16[m, :] @ W.T + bias_scale * bias   (valid rows 0..7)
// One wave per 16x16 output tile; A16 is zero-padded 16xDM so all fragment
// loads are unconditional (EXEC stays all-ones, no per-iter exec save).
// K unrolled by 16 -> 4x V_WMMA_F32_16X16X4_F32 per iteration, loads batched.
//
// Fragment layouts (ISA 7.12.2, wave32):
//   A (16x4, 2 VGPRs): row = lane%16; lanes 0-15 hold K={0,1}, lanes 16-31 K={2,3}
//   B (4x16, 2 VGPRs): col = lane%16; lanes 0-15 hold K={0,1}, lanes 16-31 K={2,3}
//   C/D (16x16, 8 VGPRs): VGPR m: lanes 0-15 -> row m, lanes 16-31 -> row m+8
// ---------------------------------------------------------------------------
__device__ __forceinline__ v2f ld2(const float* __restrict__ p) {
    float2 t = *(const float2*)p;
    v2f r; r.x = t.x; r.y = t.y;
    return r;
}

__global__ void gemm8xK_wmma(const float* __restrict__ A16,  // [16, DM], rows 8..15 zero
                             const float* __restrict__ W,    // [DM, DM], row-major
                             const float* __restrict__ bias, // [DM]
                             float bias_scale,
                             float* __restrict__ out,        // [out_rows, DM]
                             int out_rows) {                 // 8 or 16 (padded)
    const int lane  = threadIdx.x & 31;
    const int wave  = threadIdx.x >> 5;
    const int tile  = blockIdx.x * (blockDim.x >> 5) + wave; // 0..63
    const int colBase = tile * 16;
    const int rc    = lane & 15;            // A-row / B-col for this lane
    const int khalf = (lane >> 4) << 1;     // 0 or 2

    const float* __restrict__ aRow = A16 + (size_t)rc * DM + khalf;
    const float* __restrict__ wRow = W + (size_t)(colBase + rc) * DM + khalf;

    v8f c = {};
    for (int k = 0; k < DM; k += 16) {
        v2f a0 = ld2(aRow + k);
        v2f a1 = ld2(aRow + k + 4);
        v2f a2 = ld2(aRow + k + 8);
        v2f a3 = ld2(aRow + k + 12);
        v2f b0 = ld2(wRow + k);
        v2f b1 = ld2(wRow + k + 4);
        v2f b2 = ld2(wRow + k + 8);
        v2f b3 = ld2(wRow + k + 12);
        c = __builtin_amdgcn_wmma_f32_16x16x4_f32(false, a0, false, b0, (short)0, c, false, false);
        c = __builtin_amdgcn_wmma_f32_16x16x4_f32(false, a1, false, b1, (short)0, c, false, false);
        c = __builtin_amdgcn_wmma_f32_16x16x4_f32(false, a2, false, b2, (short)0, c, false, false);
        c = __builtin_amdgcn_wmma_f32_16x16x4_f32(false, a3, false, b3, (short)0, c, false, false);
    }

    const int col = colBase + rc;
    if (lane < 16) {
        const float bb = bias_scale * bias[col];
        #pragma unroll
        for (int m = 0; m < NB; ++m)
            out[(size_t)m * DM + col] = c[m] + bb;
    } else if (out_rows == 16) {
        // zero padding rows 8..15 for the next GEMM's A
        #pragma unroll
        for (int m = 0; m < NB; ++m)
            out[(size_t)(m + NB) * DM + col] = 0.0f;
    }
}

// ---------------------------------------------------------------------------
// Launch
// ---------------------------------------------------------------------------
extern "C" void kernel_launch(void* const* d_in, const int* in_sizes, int n_in,
                              void* d_out, int out_size, void* d_ws, size_t ws_size,
                              hipStream_t stream) {
    // setup_inputs order: Q K V Wq bq Wk bk Wv bv Wo bo
    const float* V  = (const float*)d_in[2];
    const float* Wv = (const float*)d_in[7];
    const float* bv = (const float*)d_in[8];
    const float* Wo = (const float*)d_in[9];
    const float* bo = (const float*)d_in[10];
    float* out = (float*)d_out;

    float* ws      = (float*)d_ws;
    float* partial = ws;                              // 8*16*1024 f = 512 KB
    float* vsum16  = partial + NB * SCHUNKS * DM;     // 16*1024  f = 64 KB
    float* u16     = vsum16 + 16 * DM;                // 16*1024  f = 64 KB

    // 1) partial sums over S (128 blocks x 8 waves, b128 streams)
    reduceV_partial<<<dim3(1, SCHUNKS, NB), 256, 0, stream>>>(V, partial);

    // 2) combine -> zero-padded Vsum16
    combine_partials<<<(16 * DM) / 256, 256, 0, stream>>>(partial, vsum16);

    // 3) u16 = Vsum16 @ Wv.T + SEQ * bv (rows 8..15 zeroed)
    gemm8xK_wmma<<<8, 256, 0, stream>>>(vsum16, Wv, bv, (float)SEQ, u16, 16);

    // 4) out = u16 @ Wo.T + bo (8 valid rows only)
    gemm8xK_wmma<<<8, 256, 0, stream>>>(u16, Wo, bo, 1.0f, out, 8);
}